// GLCN_31147102830954
// MI455X (gfx1250) — compile-verified
//
#include <hip/hip_runtime.h>
#include <math.h>

// GLCN forward on gfx1250 (MI455X, wave32, WMMA).
// Strategy: everything stays f32 (matches reference math); matrix ops use
// V_WMMA_F32_16X16X4_F32. Whole working set (~40MB) is L2-resident (192MB L2),
// so N^2 softmax/mask passes are cheap; dominant FLOPs (att@Wh, 2x ~1.07GFLOP)
// go through the tensor pipe.

#define NN      2048
#define F_INN   64
#define GG      128
#define F_OBSN  32
#define TAUf    0.1f
#define EPSf    1e-10f
#define NSLOPE  0.2f
#define NEGFILL -9e15f

typedef __attribute__((ext_vector_type(2))) float v2f;
typedef __attribute__((ext_vector_type(8))) float v8f;

// ---------------------------------------------------------------------------
// Kernel 1: A[i,j] = sigmoid((sum_k |h_i[k]-h_j[k]|*a_link[k] + gumbel)/tau),
// diagonal forced to 1.  16x16 tile per 256-thread block, h rows staged in LDS.
// ---------------------------------------------------------------------------
__global__ void build_A(const float* __restrict__ X, const float* __restrict__ gu,
                        const float* __restrict__ a_link, float* __restrict__ A_out)
{
    __shared__ float hi_s[16][F_OBSN];
    __shared__ float hj_s[16][F_OBSN];
    __shared__ float al_s[F_OBSN];
    const int i0 = blockIdx.y * 16, j0 = blockIdx.x * 16;
    const int tid = threadIdx.x; // 256 threads
    for (int s = tid; s < 16 * F_OBSN; s += 256) {
        const int r = s >> 5, c = s & 31;
        hi_s[r][c] = X[(size_t)(i0 + r) * F_INN + c]; // h = X[:, :32]
        hj_s[r][c] = X[(size_t)(j0 + r) * F_INN + c];
    }
    if (tid < F_OBSN) al_s[tid] = a_link[tid];
    __syncthreads();

    const int ti = tid >> 4, tj = tid & 15;
    float logit = 0.f;
#pragma unroll
    for (int k = 0; k < F_OBSN; ++k)
        logit += fabsf(hi_s[ti][k] - hj_s[tj][k]) * al_s[k];

    const int i = i0 + ti, j = j0 + tj;
    const float u  = gu[(size_t)i * NN + j];
    const float gn = -__logf(-__logf(u + EPSf) + EPSf);
    const float x  = (logit + gn) / TAUf;
    float a = 1.f / (1.f + __expf(-x));
    if (i == j) a = 1.f;
    A_out[(size_t)i * NN + j] = a;
}

// ---------------------------------------------------------------------------
// Kernel 2: generic f32 WMMA GEMM  Out[2048,128] = Hm[2048,K] @ W[K,128].
// One wave per 16x16 output tile; K-loop of 16x16x4 f32 WMMAs.
// A-frag layout (ISA 16x4 f32): lanes 0-15 hold K=0,1; lanes 16-31 hold K=2,3.
// ---------------------------------------------------------------------------
__global__ void gemm_f32_wmma(const float* __restrict__ Hm, const float* __restrict__ W,
                              float* __restrict__ Out, int K)
{
    const int i0   = blockIdx.x * 16;
    const int c0   = blockIdx.y * 16;
    const int lane = threadIdx.x & 31;
    const int l    = lane & 15;
    const int kh   = (lane >> 4) << 1; // 0 or 2
    v8f acc = {};
    const float* hrow = Hm + (size_t)(i0 + l) * K + kh; // 8B aligned (kh even)
    for (int kk = 0; kk < K; kk += 4) {
        v2f a = *(const v2f*)(hrow + kk);               // global_load_b64
        v2f b;
        b.x = W[(size_t)(kk + kh)     * GG + c0 + l];
        b.y = W[(size_t)(kk + kh + 1) * GG + c0 + l];
        acc = __builtin_amdgcn_wmma_f32_16x16x4_f32(false, a, false, b,
                                                    (short)0, acc, false, false);
    }
    const int hi = (lane >> 4) << 3;
#pragma unroll
    for (int r = 0; r < 8; ++r)
        Out[(size_t)(i0 + r + hi) * GG + c0 + l] = acc[r];
}

// ---------------------------------------------------------------------------
// Kernel 3: wh1 = q @ a[:G], wh2 = v @ a[G:]  (one thread per row; L2-resident)
// ---------------------------------------------------------------------------
__global__ void attn_vec(const float* __restrict__ q, const float* __restrict__ v,
                         const float* __restrict__ a_att,
                         float* __restrict__ wh1, float* __restrict__ wh2)
{
    const int row = blockIdx.x * blockDim.x + threadIdx.x;
    if (row >= NN) return;
    float s1 = 0.f, s2 = 0.f;
#pragma unroll 8
    for (int c = 0; c < GG; ++c) {
        s1 += q[(size_t)row * GG + c] * a_att[c];
        s2 += v[(size_t)row * GG + c] * a_att[GG + c];
    }
    wh1[row] = s1; wh2[row] = s2;
}

// ---------------------------------------------------------------------------
// Kernel 4: per-row masked-softmax, un-normalized.  score = A>0.01 ?
// leaky_relu(wh1[i]*wh2[j]) : -9e15.  Stores exp(score-max) and 1/rowsum.
// One 256-thread block per row. Diagonal guarantees max > NEGFILL.
// ---------------------------------------------------------------------------
__global__ void att_softmax(const float* __restrict__ A, const float* __restrict__ wh1,
                            const float* __restrict__ wh2, float* __restrict__ att,
                            float* __restrict__ rinv)
{
    const int i   = blockIdx.x;
    const int tid = threadIdx.x;
    __shared__ float red[256];
    const float w1 = wh1[i];

    float m = -INFINITY;
    for (int j = tid; j < NN; j += 256) {
        float e = w1 * wh2[j];
        e = e > 0.f ? e : NSLOPE * e;
        const float s = (A[(size_t)i * NN + j] > 0.01f) ? e : NEGFILL;
        m = fmaxf(m, s);
    }
    red[tid] = m; __syncthreads();
    for (int s = 128; s > 0; s >>= 1) {
        if (tid < s) red[tid] = fmaxf(red[tid], red[tid + s]);
        __syncthreads();
    }
    m = red[0]; __syncthreads();

    float sum = 0.f;
    for (int j = tid; j < NN; j += 256) {
        float e = w1 * wh2[j];
        e = e > 0.f ? e : NSLOPE * e;
        const float s  = (A[(size_t)i * NN + j] > 0.01f) ? e : NEGFILL;
        const float ex = __expf(s - m);
        att[(size_t)i * NN + j] = ex;
        sum += ex;
    }
    red[tid] = sum; __syncthreads();
    for (int s = 128; s > 0; s >>= 1) {
        if (tid < s) red[tid] += red[tid + s];
        __syncthreads();
    }
    if (tid == 0) rinv[i] = 1.f / red[0];
}

// ---------------------------------------------------------------------------
// Kernel 5: Hout = relu( (att @ Wh) * rinv[row] ).  K = 2048, f32 WMMA,
// prefetch the att row ahead into L2/L0 (global_prefetch_b8).
// ---------------------------------------------------------------------------
__global__ void att_gemm_wmma(const float* __restrict__ att, const float* __restrict__ Wh,
                              const float* __restrict__ rinv, float* __restrict__ Hout)
{
    const int i0   = blockIdx.x * 16;
    const int c0   = blockIdx.y * 16;
    const int lane = threadIdx.x & 31;
    const int l    = lane & 15;
    const int kh   = (lane >> 4) << 1;
    v8f acc = {};
    const float* arow = att + (size_t)(i0 + l) * NN + kh;
    for (int kk = 0; kk < NN; kk += 4) {
        if ((kk & 63) == 0)
            __builtin_prefetch(arow + kk + 128, 0, 1);
        v2f a = *(const v2f*)(arow + kk);
        v2f b;
        b.x = Wh[(size_t)(kk + kh)     * GG + c0 + l];
        b.y = Wh[(size_t)(kk + kh + 1) * GG + c0 + l];
        acc = __builtin_amdgcn_wmma_f32_16x16x4_f32(false, a, false, b,
                                                    (short)0, acc, false, false);
    }
    const int hi = (lane >> 4) << 3;
#pragma unroll
    for (int r = 0; r < 8; ++r) {
        const int row = i0 + r + hi;
        const float vout = acc[r] * rinv[row];
        Hout[(size_t)row * GG + c0 + l] = fmaxf(vout, 0.f);
    }
}

// ---------------------------------------------------------------------------
extern "C" void kernel_launch(void* const* d_in, const int* in_sizes, int n_in,
                              void* d_out, int out_size, void* d_ws, size_t ws_size,
                              hipStream_t stream)
{
    const float* X      = (const float*)d_in[0];
    const float* gu     = (const float*)d_in[1];
    const float* a_link = (const float*)d_in[2];
    const float* Ws0    = (const float*)d_in[3];
    const float* Ws1    = (const float*)d_in[4];
    const float* Wq0    = (const float*)d_in[5];
    const float* Wq1    = (const float*)d_in[6];
    const float* Wv0    = (const float*)d_in[7];
    const float* Wv1    = (const float*)d_in[8];
    const float* a0     = (const float*)d_in[9];
    const float* a1     = (const float*)d_in[10];

    // d_out = [H (N*G) | A (N*N) | X (N*F_IN)]
    float* out   = (float*)d_out;
    float* H_out = out;
    float* A_out = out + (size_t)NN * GG;
    float* X_out = A_out + (size_t)NN * NN;

    // workspace layout (~21MB): att | Wh | q | v | H1 | wh1 | wh2 | rinv
    float* ws   = (float*)d_ws;
    float* att  = ws;
    float* Wh   = att + (size_t)NN * NN;
    float* qb   = Wh  + (size_t)NN * GG;
    float* vb   = qb  + (size_t)NN * GG;
    float* H1   = vb  + (size_t)NN * GG;
    float* wh1  = H1  + (size_t)NN * GG;
    float* wh2  = wh1 + NN;
    float* rinv = wh2 + NN;

    // ---- link-prediction graph A + X passthrough ----
    build_A<<<dim3(NN / 16, NN / 16), 256, 0, stream>>>(X, gu, a_link, A_out);
    hipMemcpyAsync(X_out, X, (size_t)NN * F_INN * sizeof(float),
                   hipMemcpyDeviceToDevice, stream);

    // ---- hop 0 (H = X, K = 64) ----
    gemm_f32_wmma<<<dim3(NN / 16, GG / 16), 32, 0, stream>>>(X, Ws0, Wh, F_INN);
    gemm_f32_wmma<<<dim3(NN / 16, GG / 16), 32, 0, stream>>>(X, Wq0, qb, F_INN);
    gemm_f32_wmma<<<dim3(NN / 16, GG / 16), 32, 0, stream>>>(X, Wv0, vb, F_INN);
    attn_vec<<<NN / 256, 256, 0, stream>>>(qb, vb, a0, wh1, wh2);
    att_softmax<<<NN, 256, 0, stream>>>(A_out, wh1, wh2, att, rinv);
    att_gemm_wmma<<<dim3(NN / 16, GG / 16), 32, 0, stream>>>(att, Wh, rinv, H1);

    // ---- hop 1 (H = H1, K = 128) ----
    gemm_f32_wmma<<<dim3(NN / 16, GG / 16), 32, 0, stream>>>(H1, Ws1, Wh, GG);
    gemm_f32_wmma<<<dim3(NN / 16, GG / 16), 32, 0, stream>>>(H1, Wq1, qb, GG);
    gemm_f32_wmma<<<dim3(NN / 16, GG / 16), 32, 0, stream>>>(H1, Wv1, vb, GG);
    attn_vec<<<NN / 256, 256, 0, stream>>>(qb, vb, a1, wh1, wh2);
    att_softmax<<<NN, 256, 0, stream>>>(A_out, wh1, wh2, att, rinv);
    att_gemm_wmma<<<dim3(NN / 16, GG / 16), 32, 0, stream>>>(att, Wh, rinv, H_out);
}